// MeanAggregator_42477226557514
// MI455X (gfx1250) — compile-verified
//
#include <hip/hip_runtime.h>
#include <math.h>

typedef __attribute__((ext_vector_type(2))) float v2f;
typedef __attribute__((ext_vector_type(4))) float v4f;
typedef __attribute__((ext_vector_type(8))) float v8f;

#define GAT_N 2048
#define GAT_M 8192
#define GAT_F 1024
#define GAT_K 10
#define GAT_ALPHA 0.2f

// ---------------------------------------------------------------------------
// Phase 1: H = X @ W  (fp32 WMMA, 16x16x4). Each wave computes a 64x64 tile:
// 4 row-subtiles x 4 col-subtiles -> 16 v_wmma_f32_16x16x4_f32 per K-step of
// 4, fed by 12 VMEM (4 x b64 for A, 8 x b32 for B). W (4MB) is L2-resident;
// the 64-row tile halves per-FLOP W re-reads vs a 32-row tile.
// ---------------------------------------------------------------------------
__global__ __launch_bounds__(256) void gat_gemm_xw(const float* __restrict__ X,
                                                   const float* __restrict__ W,
                                                   float* __restrict__ H,
                                                   int rows) {
    const int F = GAT_F;
    const int wavesPerBlock = blockDim.x >> 5;
    const int wave = blockIdx.x * wavesPerBlock + (threadIdx.x >> 5);
    const int rowTiles = rows >> 6;                 // 64-row tiles
    const int rowTile = wave % rowTiles;
    const int colTile = wave / rowTiles;            // 64-col tiles (F/64 = 16)
    const int lane = threadIdx.x & 31;
    const int r = lane & 15;                        // row/col within subtile
    const int g = lane >> 4;                        // half-wave group
    const int i0 = rowTile << 6;
    const int j0 = colTile << 6;

    v8f c[4][4];
#pragma unroll
    for (int m = 0; m < 4; ++m)
#pragma unroll
        for (int n = 0; n < 4; ++n)
#pragma unroll
            for (int v = 0; v < 8; ++v) c[m][n][v] = 0.0f;

    const float* xr[4];
#pragma unroll
    for (int m = 0; m < 4; ++m)
        xr[m] = X + (size_t)(i0 + 16 * m + r) * F;   // A subtile m, my row

#pragma unroll 2
    for (int k0 = 0; k0 < F; k0 += 4) {
        const int ka = k0 + 2 * g;      // lanes 0-15: K=k0,k0+1 ; 16-31: K=k0+2,k0+3
        v2f a[4];
#pragma unroll
        for (int m = 0; m < 4; ++m)
            a[m] = *(const v2f*)(xr[m] + ka);        // A[row][ka], A[row][ka+1]

        const float* w0 = W + (size_t)ka * F + j0 + r;  // B row ka, my column
        const float* w1 = w0 + F;                        // B row ka+1
        v2f b[4];
#pragma unroll
        for (int n = 0; n < 4; ++n) {
            b[n].x = w0[16 * n];
            b[n].y = w1[16 * n];
        }
#pragma unroll
        for (int m = 0; m < 4; ++m)
#pragma unroll
            for (int n = 0; n < 4; ++n)
                c[m][n] = __builtin_amdgcn_wmma_f32_16x16x4_f32(
                    false, a[m], false, b[n], (short)0, c[m][n], false, false);
    }

    // C/D layout: VGPR v, lanes 0-15 -> M=v ; lanes 16-31 -> M=v+8 ; N=r
#pragma unroll
    for (int m = 0; m < 4; ++m)
#pragma unroll
        for (int n = 0; n < 4; ++n)
#pragma unroll
            for (int v = 0; v < 8; ++v) {
                const int row = i0 + 16 * m + 8 * g + v;
                const int col = j0 + 16 * n + r;
                H[(size_t)row * F + col] = c[m][n][v];
            }
}

// ---------------------------------------------------------------------------
// Phase 2: s1[i] = hf[i,:] . a[:F] ; s2[j] = he[j,:] . a[F:2F]
// One wave per row, shfl reduction. hf/he are L2-resident (40MB << 192MB).
// ---------------------------------------------------------------------------
__global__ __launch_bounds__(256) void gat_rowdot(const float* __restrict__ hf,
                                                  const float* __restrict__ he,
                                                  const float* __restrict__ a,
                                                  float* __restrict__ s1,
                                                  float* __restrict__ s2) {
    const int F = GAT_F;
    const int wave = blockIdx.x * (blockDim.x >> 5) + (threadIdx.x >> 5);
    const int lane = threadIdx.x & 31;

    const float* h;
    const float* av;
    float* out;
    int row;
    if (wave < GAT_N) { h = hf; av = a;      out = s1; row = wave; }
    else              { h = he; av = a + F;  out = s2; row = wave - GAT_N; }

    const float* hr = h + (size_t)row * F;
    float acc = 0.0f;
    for (int k = lane; k < F; k += 32) acc += hr[k] * av[k];
#pragma unroll
    for (int off = 16; off > 0; off >>= 1) acc += __shfl_down(acc, off, 32);
    if (lane == 0) out[row] = acc;
}

// ---------------------------------------------------------------------------
// Phase 3: per output row i — dedupe neighbor indices (mask .set() is
// idempotent), leaky-ReLU, stable softmax over the unique set (all non-
// neighbor columns underflow to exactly 0 in fp32), then weighted gather
// of he rows. One block (256 threads) per row; float4 gather.
// ---------------------------------------------------------------------------
__global__ __launch_bounds__(256) void gat_attn_gather(const float* __restrict__ he,
                                                       const float* __restrict__ s1,
                                                       const float* __restrict__ s2,
                                                       const int* __restrict__ nidx,
                                                       float* __restrict__ out) {
    const int F = GAT_F;
    const int i = blockIdx.x;

    __shared__ int   sh_idx[GAT_K];
    __shared__ float sh_w[GAT_K];
    __shared__ int   sh_cnt;

    if (threadIdx.x == 0) {
        int   idx[GAT_K];
        float ev[GAT_K];
        int cnt = 0;
        const float s1i = s1[i];
        for (int k = 0; k < GAT_K; ++k) {
            const int j = nidx[i * GAT_K + k];
            bool dup = false;
            for (int p = 0; p < cnt; ++p) dup = dup || (idx[p] == j);
            if (!dup) {
                float e = s1i + s2[j];
                e = (e > 0.0f) ? e : GAT_ALPHA * e;   // leaky relu
                idx[cnt] = j;
                ev[cnt] = e;
                ++cnt;
            }
        }
        float mx = ev[0];
        for (int p = 1; p < cnt; ++p) mx = fmaxf(mx, ev[p]);
        float sum = 0.0f;
        for (int p = 0; p < cnt; ++p) { ev[p] = expf(ev[p] - mx); sum += ev[p]; }
        const float inv = 1.0f / sum;
        for (int p = 0; p < cnt; ++p) { sh_idx[p] = idx[p]; sh_w[p] = ev[p] * inv; }
        sh_cnt = cnt;
    }
    __syncthreads();

    const int cnt = sh_cnt;
    const int c = threadIdx.x * 4;        // 256 threads * 4 floats = 1024 = F
    v4f acc; acc[0] = 0.f; acc[1] = 0.f; acc[2] = 0.f; acc[3] = 0.f;
    for (int p = 0; p < cnt; ++p) {
        const float w = sh_w[p];
        const v4f hv = *(const v4f*)(he + (size_t)sh_idx[p] * F + c);
        acc += hv * w;
    }
    *(v4f*)(out + (size_t)i * F + c) = acc;
}

// ---------------------------------------------------------------------------
extern "C" void kernel_launch(void* const* d_in, const int* in_sizes, int n_in,
                              void* d_out, int out_size, void* d_ws, size_t ws_size,
                              hipStream_t stream) {
    const int N = GAT_N, M = GAT_M, F = GAT_F;

    const float* feat  = (const float*)d_in[0];
    const float* embed = (const float*)d_in[1];
    const float* W     = (const float*)d_in[2];
    const float* a     = (const float*)d_in[3];
    const int*   nidx  = (const int*)d_in[4];
    float* out = (float*)d_out;

    char* ws = (char*)d_ws;
    float* hf = (float*)ws;                                         // N*F
    float* he = (float*)(ws + (size_t)N * F * sizeof(float));       // M*F
    float* s1 = (float*)(ws + (size_t)(N + M) * F * sizeof(float)); // N
    float* s2 = s1 + N;                                             // M

    // hf = feat @ W : (N/64)*(F/64) waves, 8 waves/block
    {
        const int waves = (N / 64) * (F / 64);
        gat_gemm_xw<<<waves / 8, 256, 0, stream>>>(feat, W, hf, N);
    }
    // he = embed @ W
    {
        const int waves = (M / 64) * (F / 64);
        gat_gemm_xw<<<waves / 8, 256, 0, stream>>>(embed, W, he, M);
    }
    // s1, s2 : one wave per row, 8 rows per block
    gat_rowdot<<<(N + M) / 8, 256, 0, stream>>>(hf, he, a, s1, s2);

    // sparse softmax + gather : one block per output row
    gat_attn_gather<<<N, 256, 0, stream>>>(he, s1, s2, nidx, out);
}